// GridEncoder_71975061946542
// MI455X (gfx1250) — compile-verified
//
#include <hip/hip_runtime.h>

// Instant-NGP style multiresolution hash grid encoding.
// D=3, L=16, F=2, N_MIN=16, per_level_scale=2.0, LOG2_T=19, BOUND=1.0
//
// Level table offsets (entries, not floats), precomputed from the reference:
//   l0: dense 17^3 = 4913     l1: dense 33^3 = 35937   l2: dense 65^3 = 274625
//   l3..l15: hashed, 2^19 = 524288 entries each (mask 0x7FFFF)

#define BLOCK 256
#define STRIDE 34  // floats per point in LDS staging (conflict-free mod 64 banks)

__global__ __launch_bounds__(BLOCK) void grid_encode_kernel(
    const float* __restrict__ inp,   // (n, 3) f32
    const float* __restrict__ emb,   // (7131219, 2) f32
    float* __restrict__ out,         // (n, 32) f32
    int n)
{
    constexpr unsigned OFF[16] = {
        0u, 4913u, 40850u, 315475u,
        839763u, 1364051u, 1888339u, 2412627u,
        2936915u, 3461203u, 3985491u, 4509779u,
        5034067u, 5558355u, 6082643u, 6606931u
    };
    constexpr unsigned P1 = 2654435761u;
    constexpr unsigned P2 = 805459861u;
    constexpr unsigned HASH_MASK = (1u << 19) - 1u;

    __shared__ float st[BLOCK * STRIDE];  // 34816 B

    const int t  = threadIdx.x;
    const int gi = blockIdx.x * BLOCK + t;

    if (gi < n) {
        // Streamed input: nontemporal, don't pollute L2 (embedding table lives there).
        const float ix = __builtin_nontemporal_load(&inp[3 * (size_t)gi + 0]);
        const float iy = __builtin_nontemporal_load(&inp[3 * (size_t)gi + 1]);
        const float iz = __builtin_nontemporal_load(&inp[3 * (size_t)gi + 2]);
        const float x = (ix + 1.0f) * 0.5f;
        const float y = (iy + 1.0f) * 0.5f;
        const float z = (iz + 1.0f) * 0.5f;

#pragma unroll
        for (int l = 0; l < 16; ++l) {
            const unsigned res = 16u << l;                 // resolution
            const float scale  = (float)res - 1.0f;        // exact in f32 up to l=15

            const float px = fmaf(x, scale, 0.5f);
            const float py = fmaf(y, scale, 0.5f);
            const float pz = fmaf(z, scale, 0.5f);
            const float fpx = floorf(px), fpy = floorf(py), fpz = floorf(pz);
            const unsigned gx = (unsigned)fpx, gy = (unsigned)fpy, gz = (unsigned)fpz;
            const float fx = px - fpx, fy = py - fpy, fz = pz - fpz;

            const float wx[2] = {1.0f - fx, fx};
            const float wy[2] = {1.0f - fy, fy};
            const float wz[2] = {1.0f - fz, fz};

            // Per-level scalar base: OFF[l] folds into SADDR of the gathers.
            const float* __restrict__ eb = emb + 2ull * (unsigned long long)OFF[l];

            float a0 = 0.0f, a1 = 0.0f;
#pragma unroll
            for (int c = 0; c < 8; ++c) {
                const unsigned bx = (unsigned)(c & 1);
                const unsigned by = (unsigned)((c >> 1) & 1);
                const unsigned bz = (unsigned)((c >> 2) & 1);
                const float w = wx[bx] * wy[by] * wz[bz];

                unsigned idx;
                if (l < 3) {
                    const unsigned R1 = res + 1u;          // 17 / 33 / 65
                    idx = (gx + bx) + (gy + by) * R1 + (gz + bz) * (R1 * R1);
                } else {
                    idx = (gx + bx) ^ ((gy + by) * P1) ^ ((gz + bz) * P2);
                    idx &= HASH_MASK;
                }
                // global_load_b64 vDst, vIdx, s[eb] scale_offset; L2-resident.
                const float2 e = *(const float2*)(eb + 2u * idx);
                a0 = fmaf(w, e.x, a0);
                a1 = fmaf(w, e.y, a1);
            }
            // Stage this level's 2 features to LDS (bank-conflict-free: 34*t mod 64
            // is a distinct even bank per lane; float2 covers even+odd pair).
            *(float2*)&st[t * STRIDE + 2 * l] = make_float2(a0, a1);
        }
    }

    __syncthreads();

    // Coalesced, nontemporal output: block writes a contiguous [BLOCK*32] chunk.
    // Within a wave: p (= point-in-block) is uniform, c runs 0..31 -> consecutive
    // addresses, and LDS banks (34p + c) mod 64 are conflict-free.
    const int nInBlock   = min(n - blockIdx.x * BLOCK, BLOCK);
    const int validElems = nInBlock * 32;
    const size_t base    = (size_t)blockIdx.x * BLOCK * 32;
#pragma unroll
    for (int k = 0; k < 32; ++k) {
        const int e = k * BLOCK + t;
        if (e < validElems) {
            const int p = e >> 5;
            const int c = e & 31;
            __builtin_nontemporal_store(st[p * STRIDE + c], &out[base + e]);
        }
    }
}

extern "C" void kernel_launch(void* const* d_in, const int* in_sizes, int n_in,
                              void* d_out, int out_size, void* d_ws, size_t ws_size,
                              hipStream_t stream)
{
    (void)n_in; (void)out_size; (void)d_ws; (void)ws_size;
    const float* inp = (const float*)d_in[0];   // (n, 3) float32
    const float* emb = (const float*)d_in[1];   // (7131219, 2) float32
    float* out = (float*)d_out;                 // (n, 32) float32

    const int n = in_sizes[0] / 3;
    const int blocks = (n + BLOCK - 1) / BLOCK;
    hipLaunchKernelGGL(grid_encode_kernel, dim3(blocks), dim3(BLOCK), 0, stream,
                       inp, emb, out, n);
}